// MHA_74577812127961
// MI455X (gfx1250) — compile-verified
//
#include <hip/hip_runtime.h>
#include <math.h>

// ---------------------------------------------------------------------------
// MHA block for B=4, T=1024, dim=1024, n_heads=16, dh=64, fp32, causal + RoPE.
// All matmuls on V_WMMA_F32_16X16X4_F32 (fp32 A/B, fp32 accum).
// Round 3: weights + V pre-transposed so every fragment load is a b64;
// 32x64 GEMM wave tile (0.75 vmem per WMMA); double-buffered K chunks.
// ---------------------------------------------------------------------------

typedef __attribute__((ext_vector_type(2))) float v2f;
typedef __attribute__((ext_vector_type(8))) float v8f;

__device__ __forceinline__ v8f wmma4(v2f a, v2f b, v8f c) {
    // D = A(16x4) * B(4x16) + C(16x16), fp32
    return __builtin_amdgcn_wmma_f32_16x16x4_f32(
        /*neg_a=*/false, a, /*neg_b=*/false, b,
        /*c_mod=*/(short)0, c, /*reuse_a=*/false, /*reuse_b=*/false);
}

// ---------------------------------------------------------------------------
// fp32 WMMA GEMM with pre-transposed B: C[M x N] = A[M x K] * Bt[N x K]^T.
// One wave computes a 32 x 64 tile of C; K in double-buffered chunks of 16.
// Fragment layouts (ISA 7.12.2, wave32):
//   A 16x4 : lane l, vgpr j -> M = l&15,  K = j + 2*(l>>4)
//   B 4x16 : lane l, vgpr j -> N = l&15,  K = j + 2*(l>>4)  (b64 from Bt row)
//   C 16x16: lane l, vgpr v -> M = v + 8*(l>>4), N = l&15
// ---------------------------------------------------------------------------
struct Chunk {
    v2f a0[4], a1[4];   // A frags, rows m0.. and m0+16..
    v2f b[16];          // B frags [kstep][ntile]
};

__global__ __launch_bounds__(32) void gemm_f32_wmma(const float* __restrict__ A,
                                                    const float* __restrict__ Bt,
                                                    float* __restrict__ C,
                                                    int N, int K) {
    const int m0   = blockIdx.x * 32;
    const int n0   = blockIdx.y * 64;
    const int lane = threadIdx.x;
    const int lm   = lane & 15;
    const int half = lane >> 4;

    v8f acc00 = {}, acc01 = {}, acc02 = {}, acc03 = {};
    v8f acc10 = {}, acc11 = {}, acc12 = {}, acc13 = {};

    const float* arow0 = A + (size_t)(m0 + lm) * K + 2 * half;
    const float* arow1 = arow0 + (size_t)16 * K;
    const float* btrow[4];
#pragma unroll
    for (int nt = 0; nt < 4; ++nt)
        btrow[nt] = Bt + (size_t)(n0 + nt * 16 + lm) * K + 2 * half;

    auto load = [&](Chunk& c, int k0) {
#pragma unroll
        for (int kk = 0; kk < 4; ++kk) {
            c.a0[kk] = *(const v2f*)(arow0 + k0 + 4 * kk);
            c.a1[kk] = *(const v2f*)(arow1 + k0 + 4 * kk);
        }
#pragma unroll
        for (int nt = 0; nt < 4; ++nt)
#pragma unroll
            for (int kk = 0; kk < 4; ++kk)
                c.b[kk * 4 + nt] = *(const v2f*)(btrow[nt] + k0 + 4 * kk);
    };
    auto comp = [&](const Chunk& c) {
#pragma unroll
        for (int kk = 0; kk < 4; ++kk) {
            acc00 = wmma4(c.a0[kk], c.b[kk * 4 + 0], acc00);
            acc01 = wmma4(c.a0[kk], c.b[kk * 4 + 1], acc01);
            acc02 = wmma4(c.a0[kk], c.b[kk * 4 + 2], acc02);
            acc03 = wmma4(c.a0[kk], c.b[kk * 4 + 3], acc03);
            acc10 = wmma4(c.a1[kk], c.b[kk * 4 + 0], acc10);
            acc11 = wmma4(c.a1[kk], c.b[kk * 4 + 1], acc11);
            acc12 = wmma4(c.a1[kk], c.b[kk * 4 + 2], acc12);
            acc13 = wmma4(c.a1[kk], c.b[kk * 4 + 3], acc13);
        }
    };

    Chunk c0, c1;
    load(c0, 0);
    for (int k0 = 0; k0 < K; k0 += 32) {
        load(c1, k0 + 16);          // fetch next chunk while computing c0
        comp(c0);
        if (k0 + 32 < K) load(c0, k0 + 32);
        comp(c1);
    }

    float* crow0 = C + (size_t)(m0 + 8 * half) * N + n0 + lm;
    float* crow1 = crow0 + (size_t)16 * N;
#pragma unroll
    for (int v = 0; v < 8; ++v) {
        float* r0 = crow0 + (size_t)v * N;
        float* r1 = crow1 + (size_t)v * N;
        r0[0] = acc00[v]; r0[16] = acc01[v]; r0[32] = acc02[v]; r0[48] = acc03[v];
        r1[0] = acc10[v]; r1[16] = acc11[v]; r1[32] = acc12[v]; r1[48] = acc13[v];
    }
}

// ---------------------------------------------------------------------------
// Simple transpose: in[R][C] -> out[C][R] (coalesced read, strided write).
// Used once per call on the weights; bandwidth-trivial.
// ---------------------------------------------------------------------------
__global__ void transpose_kernel(const float* __restrict__ in,
                                 float* __restrict__ outp, int R, int C) {
    size_t idx = (size_t)blockIdx.x * blockDim.x + threadIdx.x;
    int c = (int)(idx % C);
    int r = (int)(idx / C);
    outp[(size_t)c * R + r] = in[idx];
}

// ---------------------------------------------------------------------------
// V slice transpose: Vt[(b*16+h)*64 + d][t] = qkv[b*1024 + t][2048 + h*64 + d]
// ---------------------------------------------------------------------------
__global__ void vtranspose_kernel(const float* __restrict__ qkv,
                                  float* __restrict__ vt) {
    int bh  = blockIdx.y;                 // 0..63
    int b   = bh >> 4, h = bh & 15;
    int tid = blockIdx.x * blockDim.x + threadIdx.x;  // 0..65535
    int d   = tid & 63;
    int t   = tid >> 6;
    float v = qkv[(size_t)(b * 1024 + t) * 3072 + 2048 + (size_t)h * 64 + d];
    vt[((size_t)bh * 64 + d) * 1024 + t] = v;
}

// ---------------------------------------------------------------------------
// RoPE applied in-place to q (s=0) and k (s=1) slices of qkv[4096][3072].
// ---------------------------------------------------------------------------
__global__ void rope_kernel(float* __restrict__ qkv) {
    int tid = blockIdx.x * blockDim.x + threadIdx.x;
    int j = tid & 15;
    int h = (tid >> 4) & 15;
    int s = (tid >> 8) & 1;
    int r = tid >> 9;          // 0..4095 = b*1024 + t
    int t = r & 1023;          // position (offset = 0)

    float inv = __powf(10000.0f, -(float)(2 * j) * (1.0f / 32.0f));
    float ang = (float)t * inv;
    float sn, cn;
    __sincosf(ang, &sn, &cn);

    size_t base = (size_t)r * 3072 + (size_t)s * 1024 + (size_t)h * 64 + 2 * j;
    v2f x = *(v2f*)(qkv + base);
    v2f y;
    y.x = x.x * cn - x.y * sn;
    y.y = x.y * cn + x.x * sn;
    *(v2f*)(qkv + base) = y;
}

// ---------------------------------------------------------------------------
// Causal attention, one wave per (b, h, 16-query tile).
// Two-pass online softmax; S and P*V both on V_WMMA_F32_16X16X4_F32.
// V read from transposed Vt so each V-fragment is one b64 load.
// ---------------------------------------------------------------------------
__global__ __launch_bounds__(32) void attn_kernel(const float* __restrict__ qkv,
                                                  const float* __restrict__ vt,
                                                  float* __restrict__ out) {
    const int qt   = blockIdx.x;   // 0..63 query tile
    const int h    = blockIdx.y;   // 0..15 head
    const int b    = blockIdx.z;   // 0..3  batch
    const int lane = threadIdx.x;
    const int lm   = lane & 15;
    const int half = lane >> 4;
    const int q0   = qt * 16;
    const int ldq  = 3072;

    const float* Q  = qkv + (size_t)b * 1024 * 3072 + (size_t)h * 64;
    const float* Kp = Q + 1024;
    const float* Vt = vt + (size_t)(b * 16 + h) * 64 * 1024;   // [64 d][1024 t]

    __shared__ float pbuf[16 * 18];   // stride 18: 8B-aligned float2 rows, conflict-free

    // Preload Q A-fragments for all 16 K-steps (dh=64 / 4).
    v2f qa[16];
    const float* qrow = Q + (size_t)(q0 + lm) * ldq + 2 * half;
#pragma unroll
    for (int kk = 0; kk < 16; ++kk) qa[kk] = *(const v2f*)(qrow + kk * 4);

    const float scale = 0.125f;      // 1/sqrt(64)
    float rmax[8], rsum[8];
#pragma unroll
    for (int v = 0; v < 8; ++v) { rmax[v] = -INFINITY; rsum[v] = 0.0f; }

    // ---------------- Pass 1: running row max / row sum ----------------
    for (int kt = 0; kt <= qt; ++kt) {
        const int key0 = kt * 16;
        const float* krow = Kp + (size_t)(key0 + lm) * ldq + 2 * half;

        v2f kb[16];
#pragma unroll
        for (int kk = 0; kk < 16; ++kk) kb[kk] = *(const v2f*)(krow + kk * 4);

        // warm the Vt columns this key tile will need in pass 2
        const float* vtb = Vt + key0;
        __builtin_prefetch(vtb + (size_t)lane * 1024, 0, 3);
        __builtin_prefetch(vtb + (size_t)(lane + 32) * 1024, 0, 3);

        v8f s = {};
#pragma unroll
        for (int kk = 0; kk < 16; ++kk) s = wmma4(qa[kk], kb[kk], s);

#pragma unroll
        for (int v = 0; v < 8; ++v) {
            // element: row q = q0 + v + 8*half, key = key0 + lm
            float x = s[v] * scale;
            if (kt == qt && lm > v + 8 * half) x = -INFINITY;  // causal mask
            float mx = x;
            mx = fmaxf(mx, __shfl_xor(mx, 1, 32));
            mx = fmaxf(mx, __shfl_xor(mx, 2, 32));
            mx = fmaxf(mx, __shfl_xor(mx, 4, 32));
            mx = fmaxf(mx, __shfl_xor(mx, 8, 32));
            float nm = fmaxf(rmax[v], mx);
            float e  = __expf(x - nm);
            float su = e;
            su += __shfl_xor(su, 1, 32);
            su += __shfl_xor(su, 2, 32);
            su += __shfl_xor(su, 4, 32);
            su += __shfl_xor(su, 8, 32);
            rsum[v] = rsum[v] * __expf(rmax[v] - nm) + su;
            rmax[v] = nm;
        }
    }

    float rinv[8];
#pragma unroll
    for (int v = 0; v < 8; ++v) rinv[v] = 1.0f / rsum[v];

    // ---------------- Pass 2: P = softmax(S), O += P * V ----------------
    v8f o0 = {}, o1 = {}, o2 = {}, o3 = {};
    for (int kt = 0; kt <= qt; ++kt) {
        const int key0 = kt * 16;
        const float* krow = Kp + (size_t)(key0 + lm) * ldq + 2 * half;

        v2f kb[16];
#pragma unroll
        for (int kk = 0; kk < 16; ++kk) kb[kk] = *(const v2f*)(krow + kk * 4);

        v8f s = {};
#pragma unroll
        for (int kk = 0; kk < 16; ++kk) s = wmma4(qa[kk], kb[kk], s);

#pragma unroll
        for (int v = 0; v < 8; ++v) {
            float x = s[v] * scale;
            if (kt == qt && lm > v + 8 * half) x = -INFINITY;
            float p = __expf(x - rmax[v]) * rinv[v];
            pbuf[(v + 8 * half) * 18 + lm] = p;       // D-layout -> LDS
        }

        // V B-fragments from Vt: one b64 each; overlap with LDS store drain.
        v2f vb[16];   // [kstep][ntile]
#pragma unroll
        for (int kk = 0; kk < 4; ++kk)
#pragma unroll
            for (int nt = 0; nt < 4; ++nt)
                vb[kk * 4 + nt] = *(const v2f*)(Vt + (size_t)(nt * 16 + lm) * 1024
                                                + key0 + kk * 4 + 2 * half);

        // Intra-wave cross-lane LDS dependency: wait for the P stores.
        asm volatile("s_wait_dscnt 0x0" ::: "memory");

        v2f pa[4];
#pragma unroll
        for (int kk = 0; kk < 4; ++kk)
            pa[kk] = *(const v2f*)&pbuf[lm * 18 + kk * 4 + 2 * half];

#pragma unroll
        for (int kk = 0; kk < 4; ++kk) {
            o0 = wmma4(pa[kk], vb[kk * 4 + 0], o0);
            o1 = wmma4(pa[kk], vb[kk * 4 + 1], o1);
            o2 = wmma4(pa[kk], vb[kk * 4 + 2], o2);
            o3 = wmma4(pa[kk], vb[kk * 4 + 3], o3);
        }
        // LDS ops from one wave execute in order: next tile's stores cannot
        // pass this tile's loads, so no extra wait is needed here.
    }

    // Store O (already normalized): out[b*1024 + q][h*64 + d]
    float* obase = out + (size_t)(b * 1024 + q0) * 1024 + (size_t)h * 64 + lm;
#pragma unroll
    for (int v = 0; v < 8; ++v) {
        float* r = obase + (size_t)(v + 8 * half) * 1024;
        r[0]  = o0[v];
        r[16] = o1[v];
        r[32] = o2[v];
        r[48] = o3[v];
    }
}

// ---------------------------------------------------------------------------
// Launch: W transposes -> qkv GEMM -> RoPE -> V transpose -> attention ->
// output GEMM, all on `stream`.
// Workspace (floats): qkv 12.6M | attn 4.2M | WqkvT 3.1M | WoutT 1.0M | Vt 4.2M
// = 96 MB total.
// ---------------------------------------------------------------------------
extern "C" void kernel_launch(void* const* d_in, const int* in_sizes, int n_in,
                              void* d_out, int out_size, void* d_ws, size_t ws_size,
                              hipStream_t stream) {
    (void)in_sizes; (void)n_in; (void)out_size; (void)ws_size;

    const float* x    = (const float*)d_in[0];   // [4, 1024, 1024]
    const float* Wqkv = (const float*)d_in[1];   // [1024, 3072]
    const float* Wout = (const float*)d_in[2];   // [1024, 1024]
    float* out = (float*)d_out;                  // [4, 1024, 1024]

    float* qkv   = (float*)d_ws;                         // 4096 x 3072
    float* attn  = qkv   + (size_t)4096 * 3072;          // 4096 x 1024
    float* wqkvT = attn  + (size_t)4096 * 1024;          // 3072 x 1024
    float* woutT = wqkvT + (size_t)3072 * 1024;          // 1024 x 1024
    float* vtbuf = woutT + (size_t)1024 * 1024;          // 64 x 64 x 1024

    // 0) transpose weights (once per call, bandwidth-trivial)
    transpose_kernel<<<(1024 * 3072) / 256, 256, 0, stream>>>(Wqkv, wqkvT, 1024, 3072);
    transpose_kernel<<<(1024 * 1024) / 256, 256, 0, stream>>>(Wout, woutT, 1024, 1024);

    // 1) qkv = x @ Wqkv   (M=4096, N=3072, K=1024), B pre-transposed
    dim3 g1(4096 / 32, 3072 / 64);
    gemm_f32_wmma<<<g1, 32, 0, stream>>>(x, wqkvT, qkv, 3072, 1024);

    // 2) RoPE on q and k slices, in place
    rope_kernel<<<2097152 / 256, 256, 0, stream>>>(qkv);

    // 3) transpose V slices for b64 V-fragment loads
    dim3 gv(65536 / 256, 64);
    vtranspose_kernel<<<gv, 256, 0, stream>>>(qkv, vtbuf);

    // 4) causal attention per (b, h, qtile)
    dim3 g2(64, 16, 4);
    attn_kernel<<<g2, 32, 0, stream>>>(qkv, vtbuf, attn);

    // 5) out = attn @ Wout (M=4096, N=1024, K=1024), B pre-transposed
    dim3 g3(4096 / 32, 1024 / 64);
    gemm_f32_wmma<<<g3, 32, 0, stream>>>(attn, woutT, out, 1024, 1024);
}